// region_wei_loss_28724741275888
// MI455X (gfx1250) — compile-verified
//
#include <hip/hip_runtime.h>
#include <math.h>

#define BB   8
#define CC   14
#define HH   512
#define WWD  512
#define HW   (HH * WWD)
#define NPIX (BB * HW)
#define NTILES (NPIX / 16)
#define DELTA_F  0.7f
#define FUS_WI_F 0.01f

typedef float v2f __attribute__((ext_vector_type(2)));
typedef float v8f __attribute__((ext_vector_type(8)));

// ---------------- workspace layout ----------------
// float cm[256]   : confusion matrix accumulator (14x14 used)
// float wcp[256]  : zero-padded 16x16 wc matrix (B operand for WMMA)
// int   tmin      : global min of y_true

__global__ void rwl_init_kernel(float* cm, int* tmin, float* out) {
    int tid = threadIdx.x;
    cm[tid] = 0.0f;                     // 256 threads zero 256 floats
    if (tid == 0) { *tmin = 0x7fffffff; out[0] = 0.0f; }
}

__global__ void rwl_min_kernel(const int* __restrict__ y_true, int* tmin) {
    int v = 0x7fffffff;
    for (int p = blockIdx.x * blockDim.x + threadIdx.x; p < NPIX;
         p += gridDim.x * blockDim.x)
        v = min(v, y_true[p]);
    #pragma unroll
    for (int off = 16; off >= 1; off >>= 1)
        v = min(v, __shfl_xor(v, off, 32));
    if ((threadIdx.x & 31) == 0) atomicMin(tmin, v);
}

// Pass 1: per-pixel logsumexp + argmax; weighted CE term; confusion matrix.
__global__ void rwl_pass1_kernel(const float* __restrict__ y_pred,
                                 const float* __restrict__ weight,
                                 const int* __restrict__ y_true,
                                 const int* __restrict__ backlabel,
                                 const int* __restrict__ tminp,
                                 float* cm_g, float* out) {
    __shared__ float cms[196];
    int tid = threadIdx.x;
    for (int i = tid; i < 196; i += blockDim.x) cms[i] = 0.0f;
    __syncthreads();

    const int tmin_v = *tminp;
    float ce_acc = 0.0f;

    for (int p = blockIdx.x * blockDim.x + tid; p < NPIX;
         p += gridDim.x * blockDim.x) {
        int b  = p / HW;
        int hw = p - b * HW;
        const float* xp = y_pred + (size_t)b * (CC * HW) + hw;

        float v[CC];
        #pragma unroll
        for (int c = 0; c < CC; c++) v[c] = xp[(size_t)c * HW];

        int t = y_true[p];
        if (t == 255) t = tmin_v;

        float m = v[0]; int pred = 0;
        #pragma unroll
        for (int c = 1; c < CC; c++)
            if (v[c] > m) { m = v[c]; pred = c; }   // first-max argmax

        float s = 0.0f;
        #pragma unroll
        for (int c = 0; c < CC; c++) s += __expf(v[c] - m);
        float lse = m + __logf(s);

        float xt = v[0];
        #pragma unroll
        for (int c = 1; c < CC; c++) xt = (c == t) ? v[c] : xt;

        int   blr = backlabel[p];
        float blf = (blr == 0) ? 0.4f : (float)blr;

        ce_acc += weight[t] * (lse - xt) * blf;
        atomicAdd(&cms[t * CC + pred], 1.0f);
    }

    // CE contribution (scaled by DELTA / NPIX), one atomic per wave
    ce_acc *= (DELTA_F / (float)NPIX);
    #pragma unroll
    for (int off = 16; off >= 1; off >>= 1)
        ce_acc += __shfl_xor(ce_acc, off, 32);
    if ((tid & 31) == 0) atomicAdd(out, ce_acc);

    __syncthreads();
    for (int i = tid; i < 196; i += blockDim.x)
        atomicAdd(&cm_g[i], cms[i]);
}

// wc = (wei_confus + column_normalized(cm) * FUS_WI) / (1 + FUS_WI),
// zero-padded into a 16x16 matrix (WMMA B operand, rows=channel c, cols=class k).
__global__ void rwl_wc_kernel(const float* __restrict__ cm,
                              const float* __restrict__ wei_confus,
                              float* wcp) {
    int tid = threadIdx.x;          // 256 threads
    wcp[tid] = 0.0f;
    __syncthreads();
    if (tid < 196) {
        int r = tid / CC, k = tid - r * CC;
        float colsum = 0.0f;
        #pragma unroll
        for (int rr = 0; rr < CC; rr++) colsum += cm[rr * CC + k];
        if (colsum == 0.0f) colsum = 1.0f;
        float bat = cm[r * CC + k] / colsum;
        wcp[r * 16 + k] = (wei_confus[tid] + bat * FUS_WI_F) / (1.0f + FUS_WI_F);
    }
}

// Pass 2: per 16-pixel tile, rebuild probs, proj = probs(16x16pad) @ wc(16x16)
// via 4x V_WMMA_F32_16X16X4_F32, then hinge vs one-hot, backlabel-weighted sum.
__global__ void rwl_pass2_kernel(const float* __restrict__ y_pred,
                                 const int* __restrict__ y_true,
                                 const int* __restrict__ backlabel,
                                 const float* __restrict__ wcp,
                                 const int* __restrict__ tminp,
                                 float* out) {
    const int lane = threadIdx.x & 31;
    const int half = lane >> 4;      // 0: K{0,1}, 1: K{2,3} (A layout halves)
    const int lp   = lane & 15;
    const int waveGlobal = (blockIdx.x * blockDim.x + threadIdx.x) >> 5;
    const int nWaves     = (gridDim.x * blockDim.x) >> 5;
    const int tmin_v = *tminp;

    // B operand (wc) is tile-invariant: load once.
    // VGPR0 holds rows K=4j+0 (lanes 0-15) / K=4j+2 (lanes 16-31); VGPR1 = +1.
    v2f bmat[4];
    #pragma unroll
    for (int j = 0; j < 4; j++) {
        int r0 = 4 * j + 2 * half;
        bmat[j].x = wcp[r0 * 16 + lp];
        bmat[j].y = wcp[(r0 + 1) * 16 + lp];
    }

    float contrib = 0.0f;

    for (int tile = waveGlobal; tile < NTILES; tile += nWaves) {
        int pbase = tile * 16;
        int b  = pbase / HW;
        int hw = pbase - b * HW;                 // HW % 16 == 0: no b crossing
        const float* xp = y_pred + (size_t)b * (CC * HW) + hw + lp;

        // Each lane: pixel lp, channels {4j+2*half, 4j+2*half+1}, j=0..3
        float v[8];
        #pragma unroll
        for (int j = 0; j < 4; j++) {
            int c0 = 4 * j + 2 * half;
            v[2 * j]     = (c0     < CC) ? xp[(size_t)c0 * HW]       : -INFINITY;
            v[2 * j + 1] = (c0 + 1 < CC) ? xp[(size_t)(c0 + 1) * HW] : -INFINITY;
        }

        // Cooperative logsumexp: lanes L and L+16 together hold pixel L's channels.
        float m = v[0];
        #pragma unroll
        for (int i = 1; i < 8; i++) m = fmaxf(m, v[i]);
        m = fmaxf(m, __shfl_xor(m, 16, 32));
        float s = 0.0f;
        #pragma unroll
        for (int i = 0; i < 8; i++) s += __expf(v[i] - m);
        s += __shfl_xor(s, 16, 32);
        float lse = m + __logf(s);

        // A operand: probs (padded channels -> exp(-inf)=0)
        v2f a[4];
        #pragma unroll
        for (int j = 0; j < 4; j++) {
            a[j].x = __expf(v[2 * j]     - lse);
            a[j].y = __expf(v[2 * j + 1] - lse);
        }

        v8f acc = {0.f, 0.f, 0.f, 0.f, 0.f, 0.f, 0.f, 0.f};
        #pragma unroll
        for (int j = 0; j < 4; j++)
            acc = __builtin_amdgcn_wmma_f32_16x16x4_f32(
                false, a[j], false, bmat[j], (short)0, acc, false, false);

        // Per-pixel label / backlabel (lane L holds pixel L%16's values)
        int q  = pbase + lp;
        int tv = y_true[q];
        if (tv == 255) tv = tmin_v;
        int   blr = backlabel[q];
        float blf = (blr == 0) ? 0.4f : (float)blr;

        // D layout: acc[r] = proj[pixel r + 8*half][class lp]
        #pragma unroll
        for (int r = 0; r < 8; r++) {
            int src = half ? (24 + r) : r;       // lane holding this pixel's scalars
            int   tr  = __shfl(tv,  src, 32);
            float blr_f = __shfl(blf, src, 32);
            float oh = (tr == lp) ? 1.0f : 0.0f;
            contrib += blr_f * fmaxf(oh - acc[r], 0.0f);
        }
    }

    // scale: mean over 14 classes and over NPIX pixels
    contrib *= (1.0f / (14.0f * (float)NPIX));
    #pragma unroll
    for (int off = 16; off >= 1; off >>= 1)
        contrib += __shfl_xor(contrib, off, 32);
    if (lane == 0) atomicAdd(out, contrib);
}

extern "C" void kernel_launch(void* const* d_in, const int* in_sizes, int n_in,
                              void* d_out, int out_size, void* d_ws, size_t ws_size,
                              hipStream_t stream) {
    const float* y_pred     = (const float*)d_in[0];
    const float* wei_confus = (const float*)d_in[1];
    const float* weight     = (const float*)d_in[2];
    const int*   y_true     = (const int*)d_in[3];
    const int*   backlabel  = (const int*)d_in[4];
    float* out = (float*)d_out;

    float* cm   = (float*)d_ws;
    float* wcp  = cm + 256;
    int*   tmin = (int*)(wcp + 256);

    rwl_init_kernel<<<1, 256, 0, stream>>>(cm, tmin, out);
    rwl_min_kernel<<<512, 256, 0, stream>>>(y_true, tmin);
    rwl_pass1_kernel<<<1024, 256, 0, stream>>>(y_pred, weight, y_true, backlabel,
                                               tmin, cm, out);
    rwl_wc_kernel<<<1, 256, 0, stream>>>(cm, wei_confus, wcp);
    rwl_pass2_kernel<<<1024, 256, 0, stream>>>(y_pred, y_true, backlabel, wcp,
                                               tmin, out);
}